// GAT_87892210745361
// MI455X (gfx1250) — compile-verified
//
#include <hip/hip_runtime.h>
#include <hip/hip_bf16.h>
#include <math.h>

#define NNODES 50000
#define NEDGES 800000
#define IN_FEATS 128
#define HID 64
#define NHEADS 4
#define FEAT (NHEADS * HID)   // 256
#define NCLASSES 40
#define NEG_SLOPE 0.2f

typedef float v2f __attribute__((ext_vector_type(2)));
typedef float v8f __attribute__((ext_vector_type(8)));

// ---------------------------------------------------------------------------
// GEMM: C[M x N] = A[M x K] @ B[K x N] (+ bias[N]).   K, N compile-time so
// the unrolled loads use immediate instruction offsets (no per-iteration
// 64-bit address math). One wave per 16x32 output tile: two 16x16 WMMA
// accumulators sharing the A fetch; exact fp32 V_WMMA_F32_16X16X4_F32.
// A 16x4 layout (ISA 7.12.2): lanes 0-15 hold M=lane, VGPR0/1 = K0/K1;
// lanes 16-31 hold M=lane-16, VGPR0/1 = K2/K3. B mirrored (row striped across
// lanes within a VGPR). C/D: VGPR r -> row m0 + r + 8*half, col = lane's col.
// Out-of-range rows/cols are CLAMPED for loads: a lane's B value only feeds
// that lane's own output column and A row m only feeds output row m, so the
// garbage lands exclusively in rows/cols the guarded stores never write.
// Inner loop is branch-free with full EXEC (WMMA requirement).
// ---------------------------------------------------------------------------
template <int K, int N>
__global__ void wmma_gemm_f32(const float* __restrict__ A,
                              const float* __restrict__ B,
                              const float* __restrict__ bias,
                              float* __restrict__ C, int M) {
  constexpr int tilesN = (N + 15) >> 4;
  constexpr int tilePairs = (tilesN + 1) >> 1;        // 32-wide column pairs
  const int totTiles = ((M + 15) >> 4) * tilePairs;
  const int wave = (int)((blockIdx.x * blockDim.x + threadIdx.x) >> 5);
  if (wave >= totTiles) return;                       // whole-wave uniform exit
  const int m0 = (wave / tilePairs) << 4;
  const int n0 = (wave % tilePairs) << 5;             // 32 columns per wave
  const int lane = threadIdx.x & 31;
  const int half = lane >> 4;                         // K pair (0,1) vs (2,3)
  const int lm = lane & 15;

  const int row  = m0 + lm;
  const int col0 = n0 + lm;
  const int col1 = n0 + 16 + lm;
  const int rowc  = row  < M ? row  : M - 1;          // clamped load indices
  const int col0c = col0 < N ? col0 : N - 1;
  const int col1c = col1 < N ? col1 : N - 1;

  // 32-bit offsets from uniform bases -> SADDR + voffset addressing.
  unsigned aoff = (unsigned)rowc * K + 2 * half;
  unsigned b0off = (unsigned)(2 * half) * N + col0c;
  unsigned b1off = (unsigned)(2 * half) * N + col1c;

  v8f acc0 = {0.f, 0.f, 0.f, 0.f, 0.f, 0.f, 0.f, 0.f};
  v8f acc1 = {0.f, 0.f, 0.f, 0.f, 0.f, 0.f, 0.f, 0.f};

#pragma unroll 8
  for (int k = 0; k < K; k += 4) {
    const v2f av = *(const v2f*)&A[aoff];             // global_load_b64
    const v2f bv0 = {B[b0off], B[b0off + N]};
    const v2f bv1 = {B[b1off], B[b1off + N]};
    acc0 = __builtin_amdgcn_wmma_f32_16x16x4_f32(
        false, av, false, bv0, (short)0, acc0, false, false);
    acc1 = __builtin_amdgcn_wmma_f32_16x16x4_f32(
        false, av, false, bv1, (short)0, acc1, false, false);
    aoff += 4;                                        // constant strides ->
    b0off += 4 * N;                                   // folded into immediate
    b1off += 4 * N;                                   // instruction offsets
  }

  const float badd0 = (bias && col0 < N) ? bias[col0] : 0.f;
  const float badd1 = (bias && col1 < N) ? bias[col1] : 0.f;
#pragma unroll
  for (int r = 0; r < 8; ++r) {
    const int orow = m0 + r + 8 * half;
    if (orow < M) {
      float* __restrict__ crow = C + (size_t)orow * N;
      if (col0 < N) crow[col0] = acc0[r] + badd0;
      if (col1 < N) crow[col1] = acc1[r] + badd1;
    }
  }
}

// ---------------------------------------------------------------------------
// el[n,h] = sum_d feat[n,h,d]*al[h,d];  er likewise.  One wave per (n,h).
// ---------------------------------------------------------------------------
__global__ void attn_scores(const float* __restrict__ feat,
                            const float* __restrict__ al,
                            const float* __restrict__ ar,
                            float* __restrict__ el,
                            float* __restrict__ er, int NH) {
  const int w = (int)((blockIdx.x * blockDim.x + threadIdx.x) >> 5);
  if (w >= NH) return;
  const int lane = threadIdx.x & 31;
  const int n = w >> 2;
  const int h = w & 3;
  const float* __restrict__ f = feat + (size_t)n * FEAT + h * HID;
  const float* __restrict__ L = al + h * HID;
  const float* __restrict__ R = ar + h * HID;
  float f0 = f[lane], f1 = f[lane + 32];
  float sl = f0 * L[lane] + f1 * L[lane + 32];
  float sr = f0 * R[lane] + f1 * R[lane + 32];
#pragma unroll
  for (int off = 16; off > 0; off >>= 1) {
    sl += __shfl_down(sl, off, 32);
    sr += __shfl_down(sr, off, 32);
  }
  if (lane == 0) { el[w] = sl; er[w] = sr; }
}

// O (N*FEAT) <- 0 ; MX (N*H) <- -inf ; SD (N*H) <- 0
__global__ void init_layer(float* __restrict__ O, float* __restrict__ MX,
                           float* __restrict__ SD, int n_o, int n_h) {
  const int i = blockIdx.x * blockDim.x + threadIdx.x;
  if (i < n_o) O[i] = 0.f;
  if (i < n_h) { MX[i] = -INFINITY; SD[i] = 0.f; }
}

__device__ __forceinline__ float lrelu(float v) {
  return v > 0.f ? v : NEG_SLOPE * v;
}

// Monotonic float atomic-max: positive floats via signed int max, negative
// via unsigned min (init must be -inf).
__device__ __forceinline__ void atomicMaxF(float* addr, float val) {
  const int iv = __float_as_int(val);
  if (iv >= 0) atomicMax((int*)addr, iv);
  else         atomicMin((unsigned int*)addr, (unsigned int)iv);
}

// Thread per (edge, head): segment max of leaky_relu(el[src]+er[dst]) by dst.
__global__ void edge_max(const float* __restrict__ el,
                         const float* __restrict__ er,
                         const int* __restrict__ src,
                         const int* __restrict__ dst,
                         float* __restrict__ MX, int EH) {
  const int i = blockIdx.x * blockDim.x + threadIdx.x;
  if (i >= EH) return;
  const int e = i >> 2, h = i & 3;
  const int s = src[e], d = dst[e];
  atomicMaxF(&MX[d * 4 + h], lrelu(el[s * 4 + h] + er[d * 4 + h]));
}

// Thread per (edge, head): SD[dst,h] += exp(score - MX[dst,h]).
__global__ void edge_expsum(const float* __restrict__ el,
                            const float* __restrict__ er,
                            const int* __restrict__ src,
                            const int* __restrict__ dst,
                            const float* __restrict__ MX,
                            float* __restrict__ SD, int EH) {
  const int i = blockIdx.x * blockDim.x + threadIdx.x;
  if (i >= EH) return;
  const int e = i >> 2, h = i & 3;
  const int s = src[e], d = dst[e];
  const float v = lrelu(el[s * 4 + h] + er[d * 4 + h]);
  atomicAdd(&SD[d * 4 + h], __expf(v - MX[d * 4 + h]));
}

// Wave per edge: lanes 0..3 compute alpha per head, broadcast via shuffle,
// then 8 coalesced chunks of atomicAdd(O[dst], feat[src]*alpha).
__global__ void edge_message(const float* __restrict__ feat,
                             const float* __restrict__ el,
                             const float* __restrict__ er,
                             const float* __restrict__ MX,
                             const float* __restrict__ SD,
                             const int* __restrict__ src,
                             const int* __restrict__ dst,
                             float* __restrict__ O, int E) {
  const int e = (int)((blockIdx.x * blockDim.x + threadIdx.x) >> 5);
  if (e >= E) return;                         // whole-wave uniform exit
  const int lane = threadIdx.x & 31;
  const int s = src[e], d = dst[e];
  const float* __restrict__ fs = feat + (size_t)s * FEAT;
  __builtin_prefetch(fs + lane * 8, 0, 0);    // global_prefetch_b8: warm gather row
  float alpha = 0.f;
  if (lane < 4) {
    const float v = lrelu(el[s * 4 + lane] + er[d * 4 + lane]);
    alpha = __expf(v - MX[d * 4 + lane]) / SD[d * 4 + lane];
  }
  float* __restrict__ od = O + (size_t)d * FEAT;
#pragma unroll
  for (int c = 0; c < FEAT / 32; ++c) {
    const int j = c * 32 + lane;
    const float a = __shfl(alpha, j >> 6, 32);
    atomicAdd(od + j, fs[j] * a);
  }
}

// O[i] = elu(O[i] + bias[i % C]);  C must be a power of two.
__global__ void bias_elu(float* __restrict__ O, const float* __restrict__ bias,
                         int total, int C) {
  const int i = blockIdx.x * blockDim.x + threadIdx.x;
  if (i >= total) return;
  const float v = O[i] + bias[i & (C - 1)];
  O[i] = v > 0.f ? v : (__expf(v) - 1.f);
}

// Z[n, 0:64] = mean over heads of O2[n,h,:]; Z[n, 64:128] = h_dst[n,:]
__global__ void mean_concat(const float* __restrict__ O2,
                            const float* __restrict__ hdst,
                            float* __restrict__ Z, int total) {
  const int i = blockIdx.x * blockDim.x + threadIdx.x;
  if (i >= total) return;
  const int n = i >> 6, dfeat = i & 63;
  const float* o = O2 + (size_t)n * FEAT + dfeat;
  const float m = 0.25f * (o[0] + o[64] + o[128] + o[192]);
  Z[(size_t)n * 128 + dfeat] = m;
  Z[(size_t)n * 128 + 64 + dfeat] = hdst[(size_t)n * HID + dfeat];
}

// ---------------------------------------------------------------------------
extern "C" void kernel_launch(void* const* d_in, const int* in_sizes, int n_in,
                              void* d_out, int out_size, void* d_ws, size_t ws_size,
                              hipStream_t stream) {
  (void)in_sizes; (void)n_in; (void)out_size; (void)ws_size;
  const float* x    = (const float*)d_in[0];
  const float* hdst = (const float*)d_in[1];
  const int*   src  = (const int*)d_in[2];
  const int*   dst  = (const int*)d_in[3];
  const float* W1   = (const float*)d_in[4];
  const float* al1  = (const float*)d_in[5];
  const float* ar1  = (const float*)d_in[6];
  const float* b1   = (const float*)d_in[7];
  const float* W2   = (const float*)d_in[8];
  const float* al2  = (const float*)d_in[9];
  const float* ar2  = (const float*)d_in[10];
  const float* b2   = (const float*)d_in[11];
  const float* Wfc  = (const float*)d_in[12];
  const float* bfc  = (const float*)d_in[13];
  float* out = (float*)d_out;

  // Workspace layout (floats): F[N*256] | O[N*256] | EL[N*4] | ER[N*4] | MX[N*4] | SD[N*4]
  float* F  = (float*)d_ws;
  float* O  = F + (size_t)NNODES * FEAT;
  float* EL = O + (size_t)NNODES * FEAT;
  float* ER = EL + (size_t)NNODES * NHEADS;
  float* MX = ER + (size_t)NNODES * NHEADS;
  float* SD = MX + (size_t)NNODES * NHEADS;
  float* Z  = F;  // reused for [mean(heads) , h_dst] after feat2 is consumed

  const int NH = NNODES * NHEADS;          // 200000
  const int EH = NEDGES * NHEADS;          // 3.2M
  const int NF = NNODES * FEAT;            // 12.8M

  const dim3 blk(256);
  auto waves_grid = [](int nwaves) { return dim3((unsigned)((nwaves + 7) / 8)); };
  auto thr_grid   = [](int nthr)   { return dim3((unsigned)((nthr + 255) / 256)); };

  auto gemm_tiles = [](int M, int N) {
    const int tilesN = (N + 15) >> 4;
    return ((M + 15) >> 4) * ((tilesN + 1) >> 1);
  };
  const int tiles1 = gemm_tiles(NNODES, FEAT);        // 3125 * 8 = 25000
  const int tilesF = gemm_tiles(NNODES, NCLASSES);    // 3125 * 2 = 6250

  // ---- Layer 1 ----
  wmma_gemm_f32<IN_FEATS, FEAT><<<waves_grid(tiles1), blk, 0, stream>>>(
      x, W1, nullptr, F, NNODES);
  attn_scores<<<waves_grid(NH), blk, 0, stream>>>(F, al1, ar1, EL, ER, NH);
  init_layer<<<thr_grid(NF), blk, 0, stream>>>(O, MX, SD, NF, NH);
  edge_max<<<thr_grid(EH), blk, 0, stream>>>(EL, ER, src, dst, MX, EH);
  edge_expsum<<<thr_grid(EH), blk, 0, stream>>>(EL, ER, src, dst, MX, SD, EH);
  edge_message<<<waves_grid(NEDGES), blk, 0, stream>>>(F, EL, ER, MX, SD, src, dst, O, NEDGES);
  bias_elu<<<thr_grid(NF), blk, 0, stream>>>(O, b1, NF, FEAT);

  // ---- Layer 2 ----
  wmma_gemm_f32<FEAT, FEAT><<<waves_grid(tiles1), blk, 0, stream>>>(
      O, W2, nullptr, F, NNODES);
  attn_scores<<<waves_grid(NH), blk, 0, stream>>>(F, al2, ar2, EL, ER, NH);
  init_layer<<<thr_grid(NF), blk, 0, stream>>>(O, MX, SD, NF, NH);
  edge_max<<<thr_grid(EH), blk, 0, stream>>>(EL, ER, src, dst, MX, EH);
  edge_expsum<<<thr_grid(EH), blk, 0, stream>>>(EL, ER, src, dst, MX, SD, EH);
  edge_message<<<waves_grid(NEDGES), blk, 0, stream>>>(F, EL, ER, MX, SD, src, dst, O, NEDGES);
  bias_elu<<<thr_grid(NF), blk, 0, stream>>>(O, b2, NF, FEAT);

  // ---- Head mean + concat + FC ----
  mean_concat<<<thr_grid(NNODES * HID), blk, 0, stream>>>(O, hdst, Z, NNODES * HID);
  wmma_gemm_f32<2 * HID, NCLASSES><<<waves_grid(tilesF), blk, 0, stream>>>(
      Z, Wfc, bfc, out, NNODES);
}